// Propagator_25881472926388
// MI455X (gfx1250) — compile-verified
//
#include <hip/hip_runtime.h>
#include <hip/hip_bf16.h>

// CDNA5 (gfx1250) AFQMC propagator.
// Pipeline:
//   1) pad_fields:     Fpad[32][1600] = fields zero-padded (branch-free WMMA A loads)
//   2) einsum_kernel:  M[t, ij] = Fpad[t,:] @ vhs[:, ij]    (fp32 WMMA 16x16x4, K=1600, HBM-bound,
//                      vhs streamed once with NT hint)
//   3) symscale:       V[t] = 0.5*sqrt(dt)*(M[t] + M[t]^T)  (in place)
//   4) hsym:           Hs = 0.5*(hmf + hmf^T)
//   5) 20-step scan: expm(-ts_h*Hs) then expm(i*V_t) via 6-term Taylor,
//      each term one WMMA matmul kernel (grid.z = re/im plane), CholQR every 5 steps.
//   6) final half-step expm, pack outputs + log_weight.

typedef __attribute__((ext_vector_type(2))) float v2f;
typedef __attribute__((ext_vector_type(8))) float v8f;

#define NB    400
#define NBB   160000      // 400*400
#define NF    1600
#define NTS   20
#define NTSP  32          // padded time slices (two 16-row WMMA M-tiles)
#define NCOL  80          // NUP+NDN
#define WPLANE 32000      // 400*80

static __device__ __forceinline__ v8f wmma_f32(v2f a, v2f b, v8f c) {
  // V_WMMA_F32_16X16X4_F32: D(16x16 f32) = A(16x4 f32) * B(4x16 f32) + C
  return __builtin_amdgcn_wmma_f32_16x16x4_f32(false, a, false, b, (short)0, c, false, false);
}

// ---------------- 0) zero-padded fields staging -------------------------------
__global__ void pad_fields_kernel(const float* __restrict__ fields, float* __restrict__ Fpad) {
  int tid = blockIdx.x * blockDim.x + threadIdx.x;
  if (tid >= NTSP * NF) return;
  const int t = tid / NF;
  Fpad[tid] = (t < NTS) ? fields[tid] : 0.f;
}

// ---------------- 1) M[t, ij] = sum_f fields[t,f] * vhs[f, ij] ----------------
// One wave per 16-wide ij tile; the wave carries BOTH t-tiles (t=0..15, 16..31 padded)
// so the 1 GB vhs stream is read exactly once. Inner loop is branch-free.
__global__ void einsum_kernel(const float* __restrict__ Fpad,   // [32,1600] zero-padded
                              const float* __restrict__ vhs,    // [1600,160000]
                              float* __restrict__ M) {          // [20,160000]
  const int wave = blockIdx.x * (blockDim.x >> 5) + (threadIdx.x >> 5);
  const int lane = threadIdx.x & 31;
  const int half = lane >> 4;
  const int l15  = lane & 15;
  const int ij0  = wave * 16;
  if (ij0 >= NBB) return;

  const float* a0p = Fpad + (size_t)l15 * NF + 2 * half;        // rows t = 0..15
  const float* a1p = Fpad + (size_t)(16 + l15) * NF + 2 * half; // rows t = 16..31 (zeros >=20)
  const float* bp  = vhs + (size_t)(2 * half) * NBB + ij0 + l15;

  v8f acc0 = {}; v8f acc1 = {};
  for (int f0 = 0; f0 < NF; f0 += 4) {
    // B fragment (4x16 of vhs): row K, col = l15 — single-pass stream, NT hint
    v2f b;
    b.x = __builtin_nontemporal_load(bp);
    b.y = __builtin_nontemporal_load(bp + NBB);
    // A fragments (16x4 of fields): row = t, K pair in VGPRs — unconditional b64
    v2f a0 = *(const v2f*)(a0p + f0);
    v2f a1 = *(const v2f*)(a1p + f0);
    acc0 = wmma_f32(a0, b, acc0);
    acc1 = wmma_f32(a1, b, acc1);
    bp += (size_t)4 * NBB;
  }
#pragma unroll
  for (int r = 0; r < 8; ++r) {
    const int t = r + 8 * half;
    M[(size_t)t * NBB + ij0 + l15] = acc0[r];
    const int t2 = 16 + t;
    if (t2 < NTS) M[(size_t)t2 * NBB + ij0 + l15] = acc1[r];
  }
}

// ---------------- 2) V[t] = 0.05*(M[t] + M[t]^T)  (0.5 sym * sqrt(dt)=0.1) ----
__global__ void symscale_kernel(float* __restrict__ M) {
  size_t tid = (size_t)blockIdx.x * blockDim.x + threadIdx.x;
  if (tid >= (size_t)NTS * NBB) return;
  const int t = (int)(tid / NBB);
  const int rem = (int)(tid % NBB);
  const int i = rem / NB, j = rem % NB;
  if (i > j) return;
  float* Mt = M + (size_t)t * NBB;
  const float a = Mt[i * NB + j], b = Mt[j * NB + i];
  const float v = 0.05f * (a + b);
  Mt[i * NB + j] = v;
  Mt[j * NB + i] = v;
}

// ---------------- 3) Hs = 0.5*(hmf + hmf^T) ----------------------------------
__global__ void hsym_kernel(const float* __restrict__ hmf, float* __restrict__ Hs) {
  int tid = blockIdx.x * blockDim.x + threadIdx.x;
  if (tid >= NBB) return;
  const int i = tid / NB, j = tid % NB;
  Hs[tid] = 0.5f * (hmf[i * NB + j] + hmf[j * NB + i]);
}

// ---------------- init: W_re = wfn, W_im = 0, logw = 0 ------------------------
__global__ void init_kernel(const float* __restrict__ wfn, float* __restrict__ W,
                            float* __restrict__ logw) {
  int tid = blockIdx.x * blockDim.x + threadIdx.x;
  if (tid == 0) logw[0] = 0.f;
  if (tid >= WPLANE) return;
  W[tid] = wfn[tid];
  W[WPLANE + tid] = 0.f;
}

// ---------------- Taylor term: Tout = sign*(A @ Tin_plane), W += Tout ---------
// imagOp=0 (real A):  re' = s*A@re,  im' = s*A@im
// imagOp=1 (A = i*V): re' = -s*V@im, im' = s*V@re
__global__ void taylor_kernel(const float* __restrict__ A,    // [400,400]
                              const float* __restrict__ Tin,  // [2][400*80]
                              float* __restrict__ Tout,       // [2][400*80]
                              float* __restrict__ W,          // [2][400*80]
                              float s, int imagOp) {
  const int it = blockIdx.x;          // 0..24  (row tile)
  const int ct = blockIdx.y;          // 0..4   (col tile)
  const int z  = blockIdx.z;          // 0 = re out, 1 = im out
  const int lane = threadIdx.x & 31;
  const int half = lane >> 4;
  const int l15  = lane & 15;

  const float* src;
  float sign;
  if (imagOp) { src = Tin + (z == 0 ? WPLANE : 0); sign = (z == 0) ? -s : s; }
  else        { src = Tin + (size_t)z * WPLANE;    sign = s; }

  const int i0 = it * 16, c0 = ct * 16;
  const float* ap = A + (size_t)(i0 + l15) * NB + 2 * half;
  const float* sp = src + (size_t)(2 * half) * NCOL + c0 + l15;

  v8f acc = {};
  for (int k0 = 0; k0 < NB; k0 += 4) {
    v2f a = *(const v2f*)(ap + k0);
    v2f b;
    b.x = sp[0];
    b.y = sp[NCOL];
    acc = wmma_f32(a, b, acc);
    sp += 4 * NCOL;
  }
  float* to = Tout + (size_t)z * WPLANE;
  float* wo = W + (size_t)z * WPLANE;
#pragma unroll
  for (int r = 0; r < 8; ++r) {
    const int row = i0 + r + 8 * half;
    const size_t idx = (size_t)row * NCOL + c0 + l15;
    const float v = sign * acc[r];
    to[idx] = v;
    wo[idx] += v;
  }
}

// ---------------- CholQR: G = W_s^H W_s  (complex 40x40, per spin) ------------
__global__ void gram_kernel(const float* __restrict__ W, float* __restrict__ G) {
  const int spin = blockIdx.x;
  const float* wr = W;
  const float* wi = W + WPLANE;
  for (int p = threadIdx.x; p < 1600; p += blockDim.x) {
    const int a = p / 40, b = p % 40;
    const int ca = spin * 40 + a, cb = spin * 40 + b;
    float sr = 0.f, si = 0.f;
    for (int r = 0; r < NB; ++r) {
      const float ar = wr[r * NCOL + ca], ai = wi[r * NCOL + ca];
      const float br = wr[r * NCOL + cb], bi = wi[r * NCOL + cb];
      sr += ar * br + ai * bi;           // conj(a)*b
      si += ar * bi - ai * br;
    }
    G[((size_t)spin * 1600 + p) * 2]     = sr;
    G[((size_t)spin * 1600 + p) * 2 + 1] = si;
  }
}

// Complex Cholesky G = R^H R (upper R, positive real diagonal) for both spins
// sequentially in one thread (deterministic logw accumulation, no atomics).
__global__ void chol_kernel(const float* __restrict__ G, float* __restrict__ R,
                            float* __restrict__ logw) {
  if (threadIdx.x != 0 || blockIdx.x != 0) return;
  float ld = 0.f;
  for (int spin = 0; spin < 2; ++spin) {
    const float* g = G + (size_t)spin * 3200;
    float* rm = R + (size_t)spin * 3200;
    for (int i = 0; i < 3200; ++i) rm[i] = 0.f;
    for (int j = 0; j < 40; ++j) {
      float d = g[(j * 40 + j) * 2];
      for (int k = 0; k < j; ++k) {
        const float rr = rm[(k * 40 + j) * 2], ri = rm[(k * 40 + j) * 2 + 1];
        d -= rr * rr + ri * ri;
      }
      const float rjj = sqrtf(fmaxf(d, 1e-30f));
      rm[(j * 40 + j) * 2] = rjj;
      ld += logf(rjj);
      const float inv = 1.f / rjj;
      for (int m = j + 1; m < 40; ++m) {
        float sr = g[(j * 40 + m) * 2], si = g[(j * 40 + m) * 2 + 1];
        for (int k = 0; k < j; ++k) {
          const float akr = rm[(k * 40 + j) * 2], aki = rm[(k * 40 + j) * 2 + 1]; // R[k,j]
          const float bkr = rm[(k * 40 + m) * 2], bki = rm[(k * 40 + m) * 2 + 1]; // R[k,m]
          sr -= akr * bkr + aki * bki;     // conj(R[k,j]) * R[k,m]
          si -= akr * bki - aki * bkr;
        }
        rm[(j * 40 + m) * 2]     = sr * inv;
        rm[(j * 40 + m) * 2 + 1] = si * inv;
      }
    }
  }
  logw[0] += ld;
}

// W_s := W_s * R^{-1}  (forward substitution over columns, one thread per row)
__global__ void trsm_kernel(float* __restrict__ W, const float* __restrict__ R) {
  __shared__ float rs[3200];
  const int spin = blockIdx.y;
  const float* rg = R + (size_t)spin * 3200;
  for (int i = threadIdx.x; i < 3200; i += blockDim.x) rs[i] = rg[i];
  __syncthreads();
  const int row = blockIdx.x * blockDim.x + threadIdx.x;
  if (row >= NB) return;
  float* wr = W;
  float* wi = W + WPLANE;
  float nr[40], ni[40];
  for (int j = 0; j < 40; ++j) {
    float ar = wr[row * NCOL + spin * 40 + j];
    float ai = wi[row * NCOL + spin * 40 + j];
    for (int k = 0; k < j; ++k) {
      const float rr = rs[(k * 40 + j) * 2], ri = rs[(k * 40 + j) * 2 + 1];
      ar -= nr[k] * rr - ni[k] * ri;
      ai -= nr[k] * ri + ni[k] * rr;
    }
    const float inv = 1.f / rs[(j * 40 + j) * 2];
    nr[j] = ar * inv;
    ni[j] = ai * inv;
  }
  for (int j = 0; j < 40; ++j) {
    wr[row * NCOL + spin * 40 + j] = nr[j];
    wi[row * NCOL + spin * 40 + j] = ni[j];
  }
}

// ---------------- pack outputs: [w_up interleaved][w_dn interleaved][logw] ----
__global__ void pack_kernel(const float* __restrict__ W, float* __restrict__ out) {
  int tid = blockIdx.x * blockDim.x + threadIdx.x;
  if (tid >= WPLANE) return;
  const int i = tid / NCOL, c = tid % NCOL;
  const int spin = c / 40, j = c % 40;
  const size_t base = (size_t)spin * WPLANE + 2 * ((size_t)i * 40 + j);
  out[base]     = W[tid];
  out[base + 1] = W[WPLANE + tid];
}

__global__ void logw_kernel(const float* __restrict__ fields, const float* __restrict__ enuc,
                            const float* __restrict__ logw_acc, float* __restrict__ out) {
  __shared__ float red[256];
  float s = 0.f;
  for (int i = threadIdx.x; i < NTS * NF; i += 256) {
    const float f = fields[i];
    s += f * f;
  }
  red[threadIdx.x] = s;
  __syncthreads();
  for (int o = 128; o > 0; o >>= 1) {
    if ((int)threadIdx.x < o) red[threadIdx.x] += red[threadIdx.x + o];
    __syncthreads();
  }
  if (threadIdx.x == 0) out[2 * WPLANE] = -0.5f * red[0] + enuc[0] + logw_acc[0];
}

extern "C" void kernel_launch(void* const* d_in, const int* in_sizes, int n_in,
                              void* d_out, int out_size, void* d_ws, size_t ws_size,
                              hipStream_t stream) {
  const float* fields = (const float*)d_in[0];   // [20,1600]
  const float* hmf    = (const float*)d_in[1];   // [400,400]
  const float* vhs    = (const float*)d_in[2];   // [1600,400,400]
  const float* wfn    = (const float*)d_in[3];   // [400,80]
  const float* enuc   = (const float*)d_in[4];   // [1]
  float* out = (float*)d_out;
  float* ws  = (float*)d_ws;

  float* M    = ws;                    // 20*160000 = 3,200,000 f  (becomes V after symscale)
  float* Hs   = M + (size_t)NTS * NBB; // 160,000 f
  float* W    = Hs + NBB;              // 64,000 f  ([re][im] planes)
  float* T    = W + 2 * WPLANE;        // 64,000 f
  float* T2   = T + 2 * WPLANE;        // 64,000 f
  float* G    = T2 + 2 * WPLANE;       // 6,400 f
  float* R    = G + 6400;              // 6,400 f
  float* logw = R + 6400;              // 1 f
  float* Fpad = logw + 1;              // 32*1600 = 51,200 f

  pad_fields_kernel<<<(NTSP * NF + 255) / 256, 256, 0, stream>>>(fields, Fpad);
  init_kernel<<<(WPLANE + 255) / 256, 256, 0, stream>>>(wfn, W, logw);
  hsym_kernel<<<(NBB + 255) / 256, 256, 0, stream>>>(hmf, Hs);
  einsum_kernel<<<NBB / 16 / 4, 128, 0, stream>>>(Fpad, vhs, M);
  symscale_kernel<<<(NTS * NBB + 255) / 256, 256, 0, stream>>>(M);

  // expm(op) applied to W with 6-term Taylor; per-term scale c/n.
  auto expm_apply = [&](const float* A, float c, int imagOp) {
    hipMemcpyAsync(T, W, (size_t)2 * WPLANE * sizeof(float),
                   hipMemcpyDeviceToDevice, stream);
    float* a = T;
    float* b = T2;
    for (int n = 1; n <= 6; ++n) {
      taylor_kernel<<<dim3(NB / 16, NCOL / 16, 2), 32, 0, stream>>>(
          A, a, b, W, c / (float)n, imagOp);
      float* tmp = a; a = b; b = tmp;
    }
  };

  for (int i = 0; i < NTS; ++i) {
    const float tsh = (i == 0) ? 0.005f : 0.01f;   // ts_h[0]=dt/2, interior=dt
    expm_apply(Hs, -tsh, 0);                        // exp(-ts_h * Hsym)
    expm_apply(M + (size_t)i * NBB, 1.0f, 1);       // exp(i * V_t), sqrt(dt) folded into V
    if ((i + 1) % 5 == 0) {
      gram_kernel<<<2, 256, 0, stream>>>(W, G);
      chol_kernel<<<1, 32, 0, stream>>>(G, R, logw);
      trsm_kernel<<<dim3((NB + 127) / 128, 2), 128, 0, stream>>>(W, R);
    }
  }
  expm_apply(Hs, -0.005f, 0);                       // trailing half-step, ts_h[20]=dt/2

  pack_kernel<<<(WPLANE + 255) / 256, 256, 0, stream>>>(W, out);
  logw_kernel<<<1, 256, 0, stream>>>(fields, enuc, logw, out);
}